// NonMaxSuppression_738734375657
// MI455X (gfx1250) — compile-verified
//
#include <hip/hip_runtime.h>
#include <stdint.h>

// NMS (Canny non-max suppression), 4096x4096 f32.
// Memory-bound: 192MB min traffic @ 23.3TB/s ~ 8.2us floor. No matmul structure,
// so no WMMA; the CDNA5 path used is async global->LDS DMA + s_wait_asynccnt.

#define IMG_W 4096
#define IMG_H 4096
#define TILE_W 256           // output columns per block
#define TILE_H 4             // output rows per block
#define HALO_W 264           // TILE_W + 4 left pad + 4 right pad (rows stay 16B aligned)
#define HALO_H 6             // TILE_H + 2
#define CHUNKS_PER_ROW (HALO_W / 4)              // 66 b128 chunks per halo row
#define TOTAL_CHUNKS   (HALO_H * CHUNKS_PER_ROW) // 396

typedef float v4f __attribute__((ext_vector_type(4)));  // clang vector: NT-builtin OK

// Per-lane async DMA of 16B from global memory directly into LDS (no VGPR staging).
__device__ __forceinline__ void async_load_f4_to_lds(const float* g, float* lds_generic) {
    uint32_t laddr = (uint32_t)(uintptr_t)lds_generic;  // low 32 bits of a generic
                                                        // LDS pointer == LDS byte offset
    uint64_t gaddr = (uint64_t)(uintptr_t)g;
    asm volatile("global_load_async_to_lds_b128 %0, %1, off"
                 :
                 : "v"(laddr), "v"(gaddr)
                 : "memory");
}

__device__ __forceinline__ void wait_asynccnt0() {
#if __has_builtin(__builtin_amdgcn_s_wait_asynccnt)
    __builtin_amdgcn_s_wait_asynccnt(0);
#else
    asm volatile("s_wait_asynccnt 0x0" ::: "memory");
#endif
}

__global__ __launch_bounds__(256) void nms_kernel(const float* __restrict__ img,
                                                  const float* __restrict__ theta,
                                                  float* __restrict__ out) {
    __shared__ float tile[HALO_H * HALO_W];  // 6336 bytes

    const int tx  = threadIdx.x;             // 0..63  (4 px each in x)
    const int ty  = threadIdx.y;             // 0..3   (1 row each)
    const int tid = ty * 64 + tx;            // 0..255
    const int x0  = blockIdx.x * TILE_W;
    const int y0  = blockIdx.y * TILE_H;

    // ---- Stage halo (rows y0-1..y0+4, cols x0-4..x0+259) into LDS via async DMA.
    // Clamped out-of-image chunks only ever feed border pixels, which are masked to 0.
    for (int c = tid; c < TOTAL_CHUNKS; c += 256) {
        const int hr = c / CHUNKS_PER_ROW;
        const int hc = c % CHUNKS_PER_ROW;
        int gy = y0 - 1 + hr;
        gy = gy < 0 ? 0 : (gy > IMG_H - 1 ? IMG_H - 1 : gy);
        int gx = x0 - 4 + hc * 4;
        gx = gx < 0 ? 0 : (gx > IMG_W - 4 ? IMG_W - 4 : gx);
        async_load_f4_to_lds(img + (size_t)gy * IMG_W + gx, &tile[hr * HALO_W + hc * 4]);
    }
    wait_asynccnt0();   // each wave drains its own ASYNCcnt
    __syncthreads();    // then the whole workgroup sees the full tile

    // ---- Compute 4 pixels: X = x0 + 4*tx + j, Y = y0 + ty.
    const int X0 = x0 + tx * 4;
    const int Y  = y0 + ty;
    const int hr = ty + 1;       // halo row of this output row
    const int hb = tx * 4 + 4;   // halo col of j=0 (16B-aligned in LDS)

    const float* rowU = &tile[(hr - 1) * HALO_W + hb];
    const float* rowM = &tile[(hr)     * HALO_W + hb];
    const float* rowD = &tile[(hr + 1) * HALO_W + hb];
    const v4f vu = *(const v4f*)rowU;
    const v4f vm = *(const v4f*)rowM;
    const v4f vd = *(const v4f*)rowD;
    // window index i corresponds to column X0-1+i
    const float u[6] = { rowU[-1], vu.x, vu.y, vu.z, vu.w, rowU[4] };
    const float m[6] = { rowM[-1], vm.x, vm.y, vm.z, vm.w, rowM[4] };
    const float d[6] = { rowD[-1], vd.x, vd.y, vd.z, vd.w, rowD[4] };

    // theta is single-use: non-temporal load keeps L2 for the img halo.
    const v4f th4 = __builtin_nontemporal_load(
        (const v4f*)(theta + (size_t)Y * IMG_W + X0));
    const float thv[4] = { th4.x, th4.y, th4.z, th4.w };

    float res[4];
    const bool yin = (Y >= 1) && (Y <= IMG_H - 2);
#pragma unroll
    for (int j = 0; j < 4; ++j) {
        const float c = m[j + 1];

        // Bit-exact angle quantization vs reference:
        //   deg = theta * f32(180/pi); if (deg<0) deg += 180;
        //   r = rint(deg/45) (IEEE div, round-half-even).
        //   Reference's q = r*45 with 180->0 remap reduces to integer tests on r:
        //   r in {0,4} -> m0, r==1 -> m45, r==2 -> m90, else (incl. NaN) -> m135.
        float deg = thv[j] * 57.29577951308232f;
        if (deg < 0.0f) deg += 180.0f;
        const float rf = rintf(deg / 45.0f);
        const int   ki = (int)rf;              // exact: rf is integral, small range
        const bool  fin = (rf == rf);          // NaN theta falls through to m135

        float n1, n2;
        if (fin && (ki == 0 || ki == 4)) {     // m0: (row, col+-1)
            n1 = m[j + 2]; n2 = m[j];
        } else if (fin && ki == 1) {           // m45: (row+1,col+1),(row-1,col-1)
            n1 = d[j + 2]; n2 = u[j];
        } else if (fin && ki == 2) {           // m90: (row+-1, col)
            n1 = d[j + 1]; n2 = u[j + 1];
        } else {                               // m135: (row+1,col-1),(row-1,col+1)
            n1 = d[j];     n2 = u[j + 2];
        }

        const int X = X0 + j;
        const bool interior = yin && (X >= 1) && (X <= IMG_W - 2);
        res[j] = ((c >= n1) && (c >= n2) && interior) ? c : 0.0f;
    }

    // Output is write-once: non-temporal store.
    v4f r4;
    r4.x = res[0]; r4.y = res[1]; r4.z = res[2]; r4.w = res[3];
    __builtin_nontemporal_store(r4, (v4f*)(out + (size_t)Y * IMG_W + X0));
}

extern "C" void kernel_launch(void* const* d_in, const int* in_sizes, int n_in,
                              void* d_out, int out_size, void* d_ws, size_t ws_size,
                              hipStream_t stream) {
    const float* img   = (const float*)d_in[0];
    const float* theta = (const float*)d_in[1];
    float* out = (float*)d_out;

    dim3 block(64, 4);                               // 256 threads = 8 wave32
    dim3 grid(IMG_W / TILE_W, IMG_H / TILE_H);       // 16 x 1024 blocks
    nms_kernel<<<grid, block, 0, stream>>>(img, theta, out);
}